// PoissonModel_44470091383465
// MI455X (gfx1250) — compile-verified
//
#include <hip/hip_runtime.h>
#include <hip/hip_bf16.h>
#include <math.h>

typedef float v2f __attribute__((ext_vector_type(2)));
typedef float v8f __attribute__((ext_vector_type(8)));

#define LRELU_SLOPE 0.2f
#define NN 8192
#define FF 512
#define HH 64
#define CC 10
#define DCONST (1.0f / (1.0f + 1e-10f))   // == 1.0f in fp32; kept for fidelity

// ---------------- small helpers ----------------

__global__ void k_zero(float* p, int n) {
    int i = blockIdx.x * blockDim.x + threadIdx.x;
    if (i < n) p[i] = 0.f;
}

// v1 = W^T a[:H], v2 = W^T a[H:], packed as V[f*16 + {0,1}], cols 2..15 = 0
__global__ void k_make_v(const float* __restrict__ W, const float* __restrict__ a,
                         float* __restrict__ V) {
    int f = threadIdx.x;              // <<<1, FF>>>
    float s0 = 0.f, s1 = 0.f;
    for (int hh = 0; hh < HH; ++hh) {
        float w = W[hh * FF + f];
        s0 += w * a[hh];
        s1 += w * a[HH + hh];
    }
    float* row = V + f * 16;
    row[0] = s0; row[1] = s1;
#pragma unroll
    for (int c = 2; c < 16; ++c) row[c] = 0.f;
}

// VW1[f*16 + c] = W1[c][f] for c<CC else 0   (pad 10 -> 16 cols)
__global__ void k_make_vw1(const float* __restrict__ W1, float* __restrict__ V) {
    int idx = blockIdx.x * blockDim.x + threadIdx.x;
    if (idx >= FF * 16) return;
    int c = idx & 15, f = idx >> 4;
    V[idx] = (c < CC) ? W1[c * FF + f] : 0.f;
}

// class counts among training nodes
__global__ void k_count(const int* __restrict__ labels, const int* __restrict__ idx_train,
                        float* __restrict__ cvec, int ntrain) {
    int j = threadIdx.x;              // <<<1, 512>>>
    if (j < 16) cvec[j] = 0.f;
    __syncthreads();
    if (j < ntrain) {
        int node = idx_train[j];
        atomicAdd(&cvec[labels[node]], 1.0f);
    }
}

// Db[node][c] = d * ((labels[node]==c) - count[c]/ntrain) for training nodes
__global__ void k_db_set(const int* __restrict__ labels, const int* __restrict__ idx_train,
                         const float* __restrict__ cvec, float* __restrict__ Db, int ntrain) {
    int j = blockIdx.x * blockDim.x + threadIdx.x;
    if (j >= ntrain) return;
    int node = idx_train[j];
    int lab = labels[node];
    float inv = 1.0f / (float)ntrain;
#pragma unroll
    for (int c = 0; c < CC; ++c)
        Db[node * CC + c] = DCONST * (((lab == c) ? 1.f : 0.f) - cvec[c] * inv);
}

// ---------------- WMMA core: 16-row tile of X[8192,512] @ V[512,16] ----------------

__device__ __forceinline__ v8f mma_rowtile(const float* __restrict__ X,
                                           const float* __restrict__ V, int row0) {
    int lane = threadIdx.x & 31;
    int m    = lane & 15;                  // row within tile / col of B
    int koff = (lane >> 4) << 1;           // 0 for lanes 0-15, 2 for lanes 16-31
    const float* xr = X + (size_t)(row0 + m) * FF;
    v8f acc = {};
    for (int k = 0; k < FF; k += 4) {
        if ((k & 63) == 0) __builtin_prefetch(xr + k + 128, 0, 1);
        int kk = k + koff;
        // A 16x4 f32: VGPR0 = K{0,2}, VGPR1 = K{1,3} per half-wave (ISA layout)
        float2 xa = *reinterpret_cast<const float2*>(xr + kk);
        v2f A = { xa.x, xa.y };
        // B 4x16 f32: row (fixed K) striped across lanes within a VGPR
        v2f B = { V[kk * 16 + m], V[(kk + 1) * 16 + m] };
        acc = __builtin_amdgcn_wmma_f32_16x16x4_f32(
            /*neg_a=*/false, A, /*neg_b=*/false, B,
            /*c_mod=*/(short)0, acc, /*reuse_a=*/false, /*reuse_b=*/false);
    }
    return acc;
}

// alpha_src / alpha_dst = h @ [v1 v2]  (cols 0,1 of the padded result)
__global__ void k_alpha(const float* __restrict__ h, const float* __restrict__ V,
                        float* __restrict__ as_, float* __restrict__ ad_) {
    int wave = (blockIdx.x * blockDim.x + threadIdx.x) >> 5;
    int row0 = wave * 16;
    v8f acc = mma_rowtile(h, V, row0);
    int lane = threadIdx.x & 31;
    int n  = lane & 15;
    int mh = (lane >> 4) * 8;              // C/D layout: VGPR r -> row r + 8*(lane>=16)
    if (n == 0) {
#pragma unroll
        for (int r = 0; r < 8; ++r) as_[row0 + mh + r] = acc[r];
    } else if (n == 1) {
#pragma unroll
        for (int r = 0; r < 8; ++r) ad_[row0 + mh + r] = acc[r];
    }
}

// xw = features @ W1^T  (cols 0..9 of the padded result), stored [N][10]
__global__ void k_xw(const float* __restrict__ feat, const float* __restrict__ V,
                     float* __restrict__ xw) {
    int wave = (blockIdx.x * blockDim.x + threadIdx.x) >> 5;
    int row0 = wave * 16;
    v8f acc = mma_rowtile(feat, V, row0);
    int lane = threadIdx.x & 31;
    int n  = lane & 15;
    int mh = (lane >> 4) * 8;
    if (n < CC) {
#pragma unroll
        for (int r = 0; r < 8; ++r) xw[(size_t)(row0 + mh + r) * CC + n] = acc[r];
    }
}

// ---------------- edge kernels ----------------

__global__ void k_edge_exp(const int* __restrict__ src, const int* __restrict__ dst,
                           const float* __restrict__ as_, const float* __restrict__ ad_,
                           float* __restrict__ ev, float* __restrict__ rowsum, int E) {
    int e = blockIdx.x * blockDim.x + threadIdx.x;
    if (e >= E) return;
    float x = as_[src[e]] + ad_[dst[e]];
    x = (x > 0.f) ? x : LRELU_SLOPE * x;
    float v = expf(x);
    ev[e] = v;
    atomicAdd(&rowsum[src[e]], v);
}

__global__ void k_edge_norm(const int* __restrict__ src, const float* __restrict__ ev,
                            const float* __restrict__ rowsum, float* __restrict__ wn, int E) {
    int e = blockIdx.x * blockDim.x + threadIdx.x;
    if (e >= E) return;
    wn[e] = DCONST * ev[e] / rowsum[src[e]];   // fold d = 1/(1+1e-10) into edge weight
}

// next = Db (+ xw at t==7) (+ self-loop d*prev for empty rows)
__global__ void k_iter_init(float* __restrict__ next, const float* __restrict__ Db,
                            const float* __restrict__ xw, const float* __restrict__ rowsum,
                            const float* __restrict__ prev, int addxw, int n) {
    int idx = blockIdx.x * blockDim.x + threadIdx.x;
    if (idx >= n) return;
    int i = idx / CC;
    float v = Db[idx];
    if (addxw) v += xw[idx];
    if (rowsum[i] == 0.f) v += DCONST * prev[idx];   // empty row -> normalized self-loop
    next[idx] = v;
}

// next[dst] += w * prev[src]  over all edges, C columns
__global__ void k_scatter(float* __restrict__ next, const float* __restrict__ prev,
                          const float* __restrict__ wn, const int* __restrict__ src,
                          const int* __restrict__ dst, int E) {
    int e = blockIdx.x * blockDim.x + threadIdx.x;
    if (e >= E) return;
    float w = wn[e];
    const float* up = prev + (size_t)src[e] * CC;
    float*       np = next + (size_t)dst[e] * CC;
#pragma unroll
    for (int c = 0; c < CC; ++c) atomicAdd(np + c, w * up[c]);
}

// ---------------- launch ----------------

extern "C" void kernel_launch(void* const* d_in, const int* in_sizes, int n_in,
                              void* d_out, int out_size, void* d_ws, size_t ws_size,
                              hipStream_t stream) {
    (void)n_in; (void)out_size; (void)ws_size;
    const float* h        = (const float*)d_in[0];
    const float* W        = (const float*)d_in[1];
    const float* a        = (const float*)d_in[2];
    const float* W1       = (const float*)d_in[3];
    const float* features = (const float*)d_in[4];
    const int*   ei       = (const int*)d_in[5];
    const int*   idx_tr   = (const int*)d_in[6];
    const int*   labels   = (const int*)d_in[7];

    const int E      = in_sizes[5] / 2;
    const int ntrain = in_sizes[6];
    const int* src = ei;
    const int* dst = ei + E;

    float* w      = (float*)d_ws;              // offsets in floats, 64B-aligned
    float* V      = w;                         //   8192
    float* VW1    = w + 8192;                  //   8192
    float* as_    = w + 16384;                 //   8192
    float* ad_    = w + 24576;                 //   8192
    float* rowsum = w + 32768;                 //   8192
    float* cvec   = w + 40960;                 //     16
    float* Db     = w + 40976;                 //  81920
    float* xw     = w + 122896;                //  81920
    float* ev     = w + 204816;                // 262144
    float* wn     = w + 466960;                // 262144
    float* utA    = w + 729104;                //  81920  (odd-step buffer)
    float* out    = (float*)d_out;             //  81920  (even-step buffer, final)

    const int THREADS = 256;
    k_zero<<<(NN + THREADS - 1) / THREADS, THREADS, 0, stream>>>(rowsum, NN);
    k_zero<<<(NN * CC + THREADS - 1) / THREADS, THREADS, 0, stream>>>(out, NN * CC);
    k_zero<<<(NN * CC + THREADS - 1) / THREADS, THREADS, 0, stream>>>(Db, NN * CC);

    k_make_v<<<1, FF, 0, stream>>>(W, a, V);
    k_make_vw1<<<(FF * 16) / THREADS, THREADS, 0, stream>>>(W1, VW1);
    k_count<<<1, 512, 0, stream>>>(labels, idx_tr, cvec, ntrain);

    // 512 row-tiles, 8 waves (1 tile each) per 256-thread block
    k_alpha<<<NN / 16 / 8, THREADS, 0, stream>>>(h, V, as_, ad_);
    k_xw<<<NN / 16 / 8, THREADS, 0, stream>>>(features, VW1, xw);

    k_db_set<<<(ntrain + THREADS - 1) / THREADS, THREADS, 0, stream>>>(labels, idx_tr, cvec, Db, ntrain);
    k_edge_exp<<<(E + THREADS - 1) / THREADS, THREADS, 0, stream>>>(src, dst, as_, ad_, ev, rowsum, E);
    k_edge_norm<<<(E + THREADS - 1) / THREADS, THREADS, 0, stream>>>(src, ev, rowsum, wn, E);

    for (int t = 1; t <= 10; ++t) {
        float* prev = ((t - 1) & 1) ? utA : out;   // step 0 state = out (zeroed)
        float* next = (t & 1) ? utA : out;         // t=10 lands in d_out
        k_iter_init<<<(NN * CC + THREADS - 1) / THREADS, THREADS, 0, stream>>>(
            next, Db, xw, rowsum, prev, (t == 7) ? 1 : 0, NN * CC);
        k_scatter<<<(E + THREADS - 1) / THREADS, THREADS, 0, stream>>>(next, prev, wn, src, dst, E);
    }
}